// GroupedMultivariatekCRPS_78640851190323
// MI455X (gfx1250) — compile-verified
//
#include <hip/hip_runtime.h>
#include <hip/hip_bf16.h>

// Problem constants (fixed by the reference's setup_inputs)
#define BATCH   2
#define EIN     8
#define TSTEPS  2
#define LATLON  20480
#define NVARS   32
#define NPAIR   28                         // 8 choose 2
#define LOCS    (BATCH * TSTEPS * LATLON)  // 81920
#define BLK     128
#define NBLK    (LOCS / BLK)               // 640 partials (64 | 640 for the WMMA reducer)

// ws layout (floats): [0] = sum(node_weights); [8..39] = fw/NVARS; [64..64+NBLK) = block partials
#define WS_FW_OFF       8
#define WS_PART_OFF     64

typedef float v8f __attribute__((ext_vector_type(8)));
typedef float v2f __attribute__((ext_vector_type(2)));

__device__ __forceinline__ float h15(float d) {
    float a = fabsf(d);
    return a * __fsqrt_rn(a);              // |d|^1.5
}

__device__ __forceinline__ float pow23(float s) {
    // s = sum |d|^1.5 >= 0 ; norm = s^(2/3)
    return s > 0.0f ? __powf(s, 0.66666667f) : 0.0f;
}

// ---------------------------------------------------------------------------
// Kernel 1: sum(node_weights) and fw_norm = fw / NVARS into workspace
// ---------------------------------------------------------------------------
__global__ __launch_bounds__(256) void kcrps_prep(const float* __restrict__ nw,
                                                  const float* __restrict__ fw,
                                                  float* __restrict__ ws) {
    __shared__ float red[256];
    int tid = threadIdx.x;
    float s = 0.0f;
    for (int i = tid; i < LATLON; i += 256) s += nw[i];
    red[tid] = s;
    __syncthreads();
    #pragma unroll
    for (int st = 128; st > 0; st >>= 1) {
        if (tid < st) red[tid] += red[tid + st];
        __syncthreads();
    }
    if (tid == 0) ws[0] = red[0];
    if (tid < NVARS) ws[WS_FW_OFF + tid] = fw[tid] * (1.0f / NVARS);
}

// ---------------------------------------------------------------------------
// Kernel 2: one lane per (b,t,node) location; streams 9x128B, 36 accumulators
// ---------------------------------------------------------------------------
__global__ __launch_bounds__(BLK) void kcrps_main(const float* __restrict__ preds,
                                                  const float* __restrict__ target,
                                                  const float* __restrict__ nodew,
                                                  const float* __restrict__ ws,
                                                  float* __restrict__ partials) {
    const int tid = threadIdx.x;
    const int loc = blockIdx.x * BLK + tid;          // 0 .. LOCS-1, exact

    const int bb  = loc / (TSTEPS * LATLON);
    const int rem = loc - bb * (TSTEPS * LATLON);    // t*LATLON + node
    const int node = rem % LATLON;

    const long tgt_off = ((long)bb * TSTEPS * LATLON + rem) * (long)NVARS;
    const long prd_off = ((long)bb * EIN * TSTEPS * LATLON + rem) * (long)NVARS;
    const long ESTR4   = ((long)TSTEPS * LATLON * NVARS) / 4;   // ensemble stride in float4

    const float4* __restrict__ y4  = reinterpret_cast<const float4*>(target + tgt_off);
    const float4* __restrict__ p4  = reinterpret_cast<const float4*>(preds + prd_off);
    const float4* __restrict__ fw4 = reinterpret_cast<const float4*>(ws + WS_FW_OFF);

    // Each of the 9 vectors for this location is exactly one 128B line: prefetch all.
    __builtin_prefetch(y4, 0, 0);
    #pragma unroll
    for (int e = 0; e < EIN; ++e) __builtin_prefetch(p4 + e * ESTR4, 0, 0);

    float accE[EIN];
    float accP[NPAIR];
    #pragma unroll
    for (int e = 0; e < EIN; ++e) accE[e] = 0.0f;
    #pragma unroll
    for (int p = 0; p < NPAIR; ++p) accP[p] = 0.0f;

    #pragma unroll
    for (int c = 0; c < NVARS / 4; ++c) {
        const float4 fw = fw4[c];
        float4 y = y4[c];
        y.x *= fw.x; y.y *= fw.y; y.z *= fw.z; y.w *= fw.w;

        float4 x[EIN];
        #pragma unroll
        for (int e = 0; e < EIN; ++e) {
            float4 v = p4[e * ESTR4 + c];
            v.x *= fw.x; v.y *= fw.y; v.z *= fw.z; v.w *= fw.w;
            x[e] = v;
        }

        #pragma unroll
        for (int e = 0; e < EIN; ++e) {
            accE[e] += h15(y.x - x[e].x) + h15(y.y - x[e].y)
                     + h15(y.z - x[e].z) + h15(y.w - x[e].w);
        }

        #pragma unroll
        for (int i = 0; i < EIN - 1; ++i) {
            #pragma unroll
            for (int j = i + 1; j < EIN; ++j) {
                const int p = (i * (15 - i)) / 2 + (j - i - 1);
                accP[p] += h15(x[i].x - x[j].x) + h15(x[i].y - x[j].y)
                         + h15(x[i].z - x[j].z) + h15(x[i].w - x[j].w);
            }
        }
    }

    float err = 0.0f;
    #pragma unroll
    for (int e = 0; e < EIN; ++e) err += pow23(accE[e]);
    float pr = 0.0f;
    #pragma unroll
    for (int p = 0; p < NPAIR; ++p) pr += pow23(accP[p]);

    // mean over e_in (=/8); ordered-pair sum * (-0.5/(8*7)) == -sum_{i<j}/56
    const float v = (err * (1.0f / EIN) - pr * (1.0f / 56.0f)) * nodew[node];

    // deterministic block reduction
    __shared__ float red[BLK];
    red[tid] = v;
    __syncthreads();
    #pragma unroll
    for (int st = BLK / 2; st > 0; st >>= 1) {
        if (tid < st) red[tid] += red[tid + st];
        __syncthreads();
    }
    if (tid == 0) partials[blockIdx.x] = red[0];
}

// ---------------------------------------------------------------------------
// Kernel 3: single wave32 reduces NBLK partials; WMMA does the 64-value sum
// (A = partials, B = ones -> D[m][n] = sum_k A[m][k], full f32 precision)
// ---------------------------------------------------------------------------
__global__ __launch_bounds__(32) void kcrps_finish(const float* __restrict__ ws,
                                                   float* __restrict__ out) {
    const int l = threadIdx.x;                       // 0..31, full wave, EXEC all-1s
    const float* __restrict__ partials = ws + WS_PART_OFF;

    float a0 = 0.0f, a1 = 0.0f;
    #pragma unroll
    for (int k = 0; k < NBLK / 64; ++k) {            // 640/64 = 10 strided groups
        a0 += partials[l + 64 * k];
        a1 += partials[l + 32 + 64 * k];
    }

    float s;
#if __has_builtin(__builtin_amdgcn_wmma_f32_16x16x4_f32)
    v2f A; A.x = a0; A.y = a1;                       // A: 16x4 f32 (64 values across wave)
    v2f Bm; Bm.x = 1.0f; Bm.y = 1.0f;                // B: 4x16 all-ones
    v8f C = {};
    v8f D = __builtin_amdgcn_wmma_f32_16x16x4_f32(
        /*neg_a=*/false, A, /*neg_b=*/false, Bm,
        /*c_mod=*/(short)0, C, /*reuse_a=*/false, /*reuse_b=*/false);
    // column n of D: lanes 0-15 hold rows 0-7, lanes 16-31 hold rows 8-15
    s = D[0] + D[1] + D[2] + D[3] + D[4] + D[5] + D[6] + D[7];
    s += __shfl_xor(s, 16, 32);                      // combine the two row halves
#else
    s = a0 + a1;
    #pragma unroll
    for (int m = 16; m > 0; m >>= 1) s += __shfl_xor(s, m, 32);
#endif

    if (l == 0) {
        const float sum_nw = ws[0];
        out[0] = s / (sum_nw * (float)BATCH);        // /sum(nw), mean over batch
    }
}

// ---------------------------------------------------------------------------
extern "C" void kernel_launch(void* const* d_in, const int* in_sizes, int n_in,
                              void* d_out, int out_size, void* d_ws, size_t ws_size,
                              hipStream_t stream) {
    const float* preds  = (const float*)d_in[0];     // (2,8,2,20480,32) f32
    const float* target = (const float*)d_in[1];     // (2,1,2,20480,32) f32
    const float* nodew  = (const float*)d_in[2];     // (20480,) f32
    const float* featw  = (const float*)d_in[3];     // (32,) f32
    float* out = (float*)d_out;                      // scalar f32
    float* ws  = (float*)d_ws;

    kcrps_prep<<<1, 256, 0, stream>>>(nodew, featw, ws);
    kcrps_main<<<NBLK, BLK, 0, stream>>>(preds, target, nodew, ws, ws + WS_PART_OFF);
    kcrps_finish<<<1, 32, 0, stream>>>(ws, out);
}